// FixedContextWordTagger_28939489641206
// MI455X (gfx1250) — compile-verified
//
#include <hip/hip_runtime.h>

typedef __attribute__((ext_vector_type(2))) float v2f;
typedef __attribute__((ext_vector_type(8))) float v8f;

#define NUM_TAGS  50
#define EMB_DIM   128
#define CTX       2
#define WIN       (2 * CTX + 1)          // 5
#define KTOT      (WIN * EMB_DIM)        // 640
#define ROWPAD    644                    // LDS row stride (floats); 644 % 64 == 4 -> conflict-free
#define TILE_M    16
#define TAG_TILES 4                      // 4 * 16 = 64 >= 50 tags

// Accumulate one 16-token tile: 5 shifts x 32 k-steps x 4 tag-tiles of
// v_wmma_f32_16x16x4_f32. EDGE=true adds zero-padding for out-of-range tokens.
template <bool EDGE>
__device__ __forceinline__ void accumulate_tile(
    v8f acc[TAG_TILES], const int* __restrict__ tokens,
    const float* __restrict__ emb, const float* __restrict__ Wlds,
    const int broff[TAG_TILES], int n0, int m16, int hi, int N) {
    // Prefetch the 5 window token ids up front so the gather chain overlaps
    // the previous shift's WMMA stream.
    int  tok[WIN];
    bool ok[WIN];
#pragma unroll
    for (int i = 0; i < WIN; ++i) {
        const int g = n0 + m16 + i - CTX;            // token for A row M = m16
        if (EDGE) {
            ok[i]  = (g >= 0) && (g < N);
            tok[i] = ok[i] ? tokens[g] : 0;
        } else {
            ok[i]  = true;
            tok[i] = tokens[g];
        }
    }
#pragma unroll
    for (int i = 0; i < WIN; ++i) {
        const float* ap = emb + (long)tok[i] * EMB_DIM + 2 * hi;
        // Per-tag-tile LDS base pointers: inner ds_loads become
        // base + small immediate (kk*16B <= 496B fits the DS offset field).
        const float* wp  = Wlds + i * EMB_DIM + 2 * hi;
        const float* wp0 = wp + broff[0];
        const float* wp1 = wp + broff[1];
        const float* wp2 = wp + broff[2];
        const float* wp3 = wp + broff[3];
#pragma unroll 4
        for (int kk = 0; kk < EMB_DIM / 4; ++kk) {
            // Load all operands first so one wait covers the 4 WMMAs below.
            v2f b0 = *(const v2f*)(wp0 + kk * 4);             // ds_load_b64
            v2f b1 = *(const v2f*)(wp1 + kk * 4);
            v2f b2 = *(const v2f*)(wp2 + kk * 4);
            v2f b3 = *(const v2f*)(wp3 + kk * 4);
            v2f a;
            if (EDGE) {
                a = (v2f){0.0f, 0.0f};
                if (ok[i]) a = *(const v2f*)(ap + kk * 4);
            } else {
                a = *(const v2f*)(ap + kk * 4);               // global_load_b64
            }
            acc[0] = __builtin_amdgcn_wmma_f32_16x16x4_f32(false, a, false, b0,
                                                           (short)0, acc[0], false, false);
            acc[1] = __builtin_amdgcn_wmma_f32_16x16x4_f32(false, a, false, b1,
                                                           (short)0, acc[1], false, false);
            acc[2] = __builtin_amdgcn_wmma_f32_16x16x4_f32(false, a, false, b2,
                                                           (short)0, acc[2], false, false);
            acc[3] = __builtin_amdgcn_wmma_f32_16x16x4_f32(false, a, false, b3,
                                                           (short)0, acc[3], false, false);
        }
    }
}

__global__ __launch_bounds__(256)
void tagger_wmma_kernel(const int* __restrict__ tokens,
                        const float* __restrict__ emb,
                        const float* __restrict__ Wmat,   // [50][640] row-major
                        const float* __restrict__ bias,   // [50]
                        float* __restrict__ out,          // [N][50]
                        int N) {
    extern __shared__ float Wlds[];      // [NUM_TAGS][ROWPAD]

    // ---- Stage W into LDS once per block --------------------------------
    for (int idx = threadIdx.x; idx < NUM_TAGS * KTOT; idx += blockDim.x) {
        int row = idx / KTOT;
        int col = idx - row * KTOT;
        Wlds[row * ROWPAD + col] = Wmat[idx];
    }
    __syncthreads();

    const int lane = threadIdx.x & 31;
    const int m16  = lane & 15;          // tile row (A) / tile col (B,D)
    const int hi   = lane >> 4;          // K half: {0,1} vs {2,3}
    const int wavesPerBlock = blockDim.x >> 5;
    const int gwave  = blockIdx.x * wavesPerBlock + (threadIdx.x >> 5);
    const int nwaves = gridDim.x * wavesPerBlock;
    const int ntiles = N / TILE_M;

    // Per-lane B rows / bias / validity (tile-invariant).
    int broff[TAG_TILES];
    bool valid[TAG_TILES];
    float bv[TAG_TILES];
#pragma unroll
    for (int t = 0; t < TAG_TILES; ++t) {
        int tag  = t * 16 + m16;
        valid[t] = (tag < NUM_TAGS);
        int r    = valid[t] ? tag : (NUM_TAGS - 1);   // clamped rows masked below
        broff[t] = r * ROWPAD;
        bv[t]    = valid[t] ? bias[tag] : 0.0f;
    }

    for (int tile = gwave; tile < ntiles; tile += nwaves) {
        const int n0 = tile * TILE_M;
        v8f acc[TAG_TILES] = {};

        if (tile > 0 && tile < ntiles - 1) {
            accumulate_tile<false>(acc, tokens, emb, Wlds, broff, n0, m16, hi, N);
        } else {
            accumulate_tile<true>(acc, tokens, emb, Wlds, broff, n0, m16, hi, N);
        }

        // ---- bias + log-softmax in registers ----------------------------
        // acc[t][j] = logit of token (n0 + j + 8*hi), tag (t*16 + m16)
#pragma unroll
        for (int j = 0; j < 8; ++j) {
            float x[TAG_TILES];
#pragma unroll
            for (int t = 0; t < TAG_TILES; ++t) x[t] = acc[t][j] + bv[t];

            float mx = fmaxf(fmaxf(x[0], x[1]), x[2]);
            if (valid[3]) mx = fmaxf(mx, x[3]);
#pragma unroll
            for (int m = 8; m >= 1; m >>= 1)
                mx = fmaxf(mx, __shfl_xor(mx, m, 32));   // 16-lane group reduce

            float s = __expf(x[0] - mx) + __expf(x[1] - mx) + __expf(x[2] - mx);
            s += valid[3] ? __expf(x[3] - mx) : 0.0f;
#pragma unroll
            for (int m = 8; m >= 1; m >>= 1)
                s += __shfl_xor(s, m, 32);

            const float lse = mx + __logf(s);
            float* op = out + (long)(n0 + j + 8 * hi) * NUM_TAGS + m16;
#pragma unroll
            for (int t = 0; t < TAG_TILES; ++t)
                if (valid[t]) op[t * 16] = x[t] - lse;
        }
    }
}

extern "C" void kernel_launch(void* const* d_in, const int* in_sizes, int n_in,
                              void* d_out, int out_size, void* d_ws, size_t ws_size,
                              hipStream_t stream) {
    const int*   tokens = (const int*)d_in[0];
    const float* emb    = (const float*)d_in[1];
    const float* Wmat   = (const float*)d_in[2];
    const float* bias   = (const float*)d_in[3];
    float*       out    = (float*)d_out;
    const int N = in_sizes[0];

    const int threads = 256;                              // 8 waves
    const int ntiles  = N / TILE_M;
    int blocks = 1024;                                    // grid-stride over tiles
    const int wavesPerBlock = threads / 32;
    if (blocks * wavesPerBlock > ntiles)
        blocks = (ntiles + wavesPerBlock - 1) / wavesPerBlock;
    if (blocks < 1) blocks = 1;

    const size_t shmem = (size_t)NUM_TAGS * ROWPAD * sizeof(float);  // ~126 KB (<320 KB WGP LDS)
    tagger_wmma_kernel<<<blocks, threads, shmem, stream>>>(tokens, emb, Wmat, bias, out, N);
}